// TripletLoss_59433757442262
// MI455X (gfx1250) — compile-verified
//
#include <hip/hip_runtime.h>
#include <hip/hip_bf16.h>

// Triplet margin loss on MI455X (gfx1250, wave32).
//
// d_pos^2 = |a|^2 + |p|^2 - 2*dot(a,p) + 2*eps*(sum(a)-sum(p)) + D*eps^2
// Dot products for 16 triplets/wave computed with V_WMMA_F32_16X16X4_F32
// (full f32 precision); row norms/sums accumulated inline from the same
// loaded data, so memory traffic is exactly the 3 gathered rows per triplet.
// Table (128 MB) is L2-resident (192 MB L2) -> L2-bandwidth bound.

typedef float v2f __attribute__((ext_vector_type(2)));
typedef float v8f __attribute__((ext_vector_type(8)));

#define DIM     128
#define NCHUNK  32          // DIM / 4  (K per WMMA = 4)
#define MARGIN  0.5f
#define EPSV    1e-6f

__global__ __launch_bounds__(256) void triplet_wmma_kernel(
    const float* __restrict__ emb,
    const int*   __restrict__ aidx,
    const int*   __restrict__ pidx,
    const int*   __restrict__ nidx,
    int T, int numTiles,
    float* __restrict__ partials)
{
    __shared__ float waveSums[8];
    const int lane = threadIdx.x & 31;
    const int wave = threadIdx.x >> 5;
    const int waveStride = gridDim.x * 8;

    float acc = 0.f;   // per-lane loss accumulator

    for (int tile = blockIdx.x * 8 + wave; tile < numTiles; tile += waveStride) {
        int trip = tile * 16 + (lane & 15);
        int tc   = trip < T ? trip : T - 1;           // clamp tail, mask later
        const int ra = aidx[tc];
        const int rp = pidx[tc];
        const int rn = nidx[tc];

        // f32 16x16x4 fragment: lane l supplies (row = l&15, cols 2*(l>>4)+{0,1})
        const size_t colBase = (size_t)((lane >> 4) << 1);
        const float* pa = emb + (size_t)ra * DIM + colBase;
        const float* pp = emb + (size_t)rp * DIM + colBase;
        const float* pn = emb + (size_t)rn * DIM + colBase;

        v8f accP = {};   // 16x16 of dot(a_i, p_j); diagonal wanted
        v8f accN = {};
        float na = 0.f, sa = 0.f;    // half-row |a|^2, sum(a)
        float np2 = 0.f, sp = 0.f;
        float nn2 = 0.f, sn = 0.f;

        #pragma unroll 8
        for (int t = 0; t < NCHUNK; ++t) {
            v2f av = *(const v2f*)(pa + 4 * t);
            v2f pv = *(const v2f*)(pp + 4 * t);
            v2f nv = *(const v2f*)(pn + 4 * t);
            accP = __builtin_amdgcn_wmma_f32_16x16x4_f32(
                false, av, false, pv, (short)0, accP, false, false);
            accN = __builtin_amdgcn_wmma_f32_16x16x4_f32(
                false, av, false, nv, (short)0, accN, false, false);
            na  += av.x * av.x + av.y * av.y;  sa += av.x + av.y;
            np2 += pv.x * pv.x + pv.y * pv.y;  sp += pv.x + pv.y;
            nn2 += nv.x * nv.x + nv.y * nv.y;  sn += nv.x + nv.y;
        }

        // merge half-row stats between lane l and l+16 (both hold row l&15)
        na  += __shfl_xor(na, 16, 32);
        sa  += __shfl_xor(sa, 16, 32);
        np2 += __shfl_xor(np2, 16, 32);
        sp  += __shfl_xor(sp, 16, 32);
        nn2 += __shfl_xor(nn2, 16, 32);
        sn  += __shfl_xor(sn, 16, 32);

        // Diagonal of C/D (16x16 f32 layout):
        //   diag j, j<8  -> lane j,     vgpr j
        //   diag j, j>=8 -> lane j+16,  vgpr j-8
        int sel = (lane < 8) ? lane : (lane - 24);   // valid only for owners
        float dotP = 0.f, dotN = 0.f;
        #pragma unroll
        for (int i = 0; i < 8; ++i) {
            dotP = (sel == i) ? accP[i] : dotP;
            dotN = (sel == i) ? accN[i] : dotN;
        }

        // owner lanes: 0..7 -> triplet lane; 24..31 -> triplet lane-16.
        // stats at lane l correspond to triplet l&15, which matches both cases.
        int  j     = (lane < 8) ? lane : (lane - 16);
        int  gi    = tile * 16 + j;
        bool owner = (lane < 8) || (lane >= 24);
        float valid = (owner && gi < T) ? 1.f : 0.f;

        const float k = (float)DIM * EPSV * EPSV;
        float d2p = na + np2 - 2.f * dotP + 2.f * EPSV * (sa - sp) + k;
        float d2n = na + nn2 - 2.f * dotN + 2.f * EPSV * (sa - sn) + k;
        float dp  = sqrtf(fmaxf(d2p, 0.f));
        float dn  = sqrtf(fmaxf(d2n, 0.f));
        acc += valid * fmaxf(dp - dn + MARGIN, 0.f);
    }

    // wave reduction (wave32)
    #pragma unroll
    for (int off = 16; off > 0; off >>= 1)
        acc += __shfl_xor(acc, off, 32);
    if (lane == 0) waveSums[wave] = acc;
    __syncthreads();
    if (threadIdx.x == 0) {
        float s = 0.f;
        #pragma unroll
        for (int w = 0; w < 8; ++w) s += waveSums[w];
        partials[blockIdx.x] = s;
    }
}

__global__ __launch_bounds__(256) void triplet_finalize_kernel(
    const float* __restrict__ partials, int nb, float invT,
    float* __restrict__ out)
{
    __shared__ float sm[256];
    float s = 0.f;
    for (int i = threadIdx.x; i < nb; i += 256) s += partials[i];
    sm[threadIdx.x] = s;
    __syncthreads();
    #pragma unroll
    for (int off = 128; off > 0; off >>= 1) {
        if (threadIdx.x < off) sm[threadIdx.x] += sm[threadIdx.x + off];
        __syncthreads();
    }
    if (threadIdx.x == 0) out[0] = sm[0] * invT;
}

extern "C" void kernel_launch(void* const* d_in, const int* in_sizes, int n_in,
                              void* d_out, int out_size, void* d_ws, size_t ws_size,
                              hipStream_t stream)
{
    const float* emb = (const float*)d_in[0];
    // d_in[1] = labels (unused; mining already done on host side)
    const int* a = (const int*)d_in[2];
    const int* p = (const int*)d_in[3];
    const int* n = (const int*)d_in[4];
    const int  T = in_sizes[2];

    int numTiles = (T + 15) / 16;
    int grid = (numTiles + 7) / 8;
    if (grid > 1024) grid = 1024;
    if (grid < 1)    grid = 1;

    float* partials = (float*)d_ws;   // grid floats, fully written before read

    triplet_wmma_kernel<<<grid, 256, 0, stream>>>(emb, a, p, n, T, numTiles, partials);
    triplet_finalize_kernel<<<1, 256, 0, stream>>>(partials, grid,
                                                   1.0f / (float)T, (float*)d_out);
}